// HMMPriorLoss_54271206752625
// MI455X (gfx1250) — compile-verified
//
#include <hip/hip_runtime.h>
#include <hip/hip_bf16.h>
#include <math.h>

// Problem constants (match reference)
#define BB 64
#define TT 2048
#define DD 256
#define KK 32
#define LOG2PI 1.8378770664093453f
#define EPSV   1e-4f
#define LOG2E  1.4426950408889634f
#define LN2    0.6931471805599453f

// Workspace layout (in floats)
#define W1P_OFF 0            // [128][32][2]  iv packed by d-pair
#define W2P_OFF 8192         // [128][32][2]  mu*iv packed by d-pair
#define OFF_OFF 16384        // [32]  -0.5*(D*log2pi + logdet + m2)
#define LPI_OFF 16416        // [32]  log2-domain log_softmax(init_logits)
#define LA_OFF  16448        // [32][32] log2-domain log_softmax(trans, -1)
#define LE_OFF  17472        // [B][T][K] log2-domain log_emit (16 MB)

typedef __attribute__((ext_vector_type(2))) float v2f;
typedef __attribute__((ext_vector_type(8))) float v8f;

// --- Types/casts matching the async-to-LDS builtin signature ------------
// Diagnostic showed param0 = "int __vector(4) __device__ *" i.e.
// vector_size(16) int in address_space(1); param1 is the LDS-side pointer.
typedef int v4i_raw __attribute__((vector_size(16)));
typedef __attribute__((address_space(1))) v4i_raw* gbl_v4i_p;
typedef __attribute__((address_space(3))) v4i_raw* lds_v4i_p;

static __device__ __forceinline__ gbl_v4i_p to_gbl(const void* p) {
    return (gbl_v4i_p)(unsigned long long)p;
}
static __device__ __forceinline__ lds_v4i_p to_lds(const void* p) {
    // LDS generic addr = aperture | offset; low 32 bits are the LDS offset.
    return (lds_v4i_p)(unsigned int)(unsigned long long)p;
}

#if __has_builtin(__builtin_amdgcn_global_load_async_to_lds_b128)
#define HAVE_ASYNC_LDS 1
#else
#define HAVE_ASYNC_LDS 0
#endif

#if __has_builtin(__builtin_amdgcn_s_wait_asynccnt)
#define WAIT_ASYNC(N) __builtin_amdgcn_s_wait_asynccnt(N)
#else
#define WAIT_ASYNC(N) asm volatile("s_wait_asynccnt %0" ::"i"(N) : "memory")
#endif

__global__ void hmm_zero_out(float* out) { out[0] = 0.0f; }

// ---------------------------------------------------------------------------
// Prep: packed B operands, per-state offsets, log2-domain log_pi / log_A.
// ---------------------------------------------------------------------------
__global__ void __launch_bounds__(256) hmm_prep(const float* __restrict__ init_logits,
                                                const float* __restrict__ trans,
                                                const float* __restrict__ means,
                                                const float* __restrict__ log_vars,
                                                float* __restrict__ ws) {
    float* W1p  = ws + W1P_OFF;
    float* W2p  = ws + W2P_OFF;
    float* offk = ws + OFF_OFF;
    float* lpi  = ws + LPI_OFF;
    float* lA   = ws + LA_OFF;
    const int tid = threadIdx.x;

    // Packed layout: e = dpair*64 + k*2 + i  -> W[2*dpair+i][k]
    for (int e = tid; e < 8192; e += 256) {
        int k = (e >> 1) & 31;
        int i = e & 1;
        int d = ((e >> 6) << 1) + i;
        float v  = fmaxf(__expf(log_vars[k * DD + d]), EPSV);
        float iv = 1.0f / v;
        W1p[e] = iv;
        W2p[e] = means[k * DD + d] * iv;
    }

    if (tid < KK) {
        const int k = tid;
        float ld = 0.0f, m2 = 0.0f;
        for (int d = 0; d < DD; ++d) {
            float v  = fmaxf(__expf(log_vars[k * DD + d]), EPSV);
            float mu = means[k * DD + d];
            ld += __logf(v);
            m2 += mu * mu / v;
        }
        offk[k] = -0.5f * ((float)DD * LOG2PI + ld + m2);

        // log_pi in log2 domain
        float m = -INFINITY;
        for (int j = 0; j < KK; ++j) m = fmaxf(m, init_logits[j]);
        float s = 0.0f;
        for (int j = 0; j < KK; ++j) s += __expf(init_logits[j] - m);
        lpi[k] = (init_logits[k] - (m + __logf(s))) * LOG2E;

        // row k of log_A in log2 domain
        float mr = -INFINITY;
        for (int c = 0; c < KK; ++c) mr = fmaxf(mr, trans[k * KK + c]);
        float sr = 0.0f;
        for (int c = 0; c < KK; ++c) sr += __expf(trans[k * KK + c] - mr);
        float lse = mr + __logf(sr);
        for (int c = 0; c < KK; ++c)
            lA[k * KK + c] = (trans[k * KK + c] - lse) * LOG2E;
    }
}

// ---------------------------------------------------------------------------
// Emission GEMM via V_WMMA_F32_16X16X4_F32, async-LDS double-buffered.
//   le2[bt,k] = log2e * (off[k] - 0.5*sum_d z^2*iv + sum_d z*(mu*iv))
// Block: 4 waves, 4 consecutive 32-row tiles, 2 LDS buffers.
// ---------------------------------------------------------------------------
#define MT 32
#define LDSPITCH 260          // float4-aligned, conflict-free A reads
#define TILES_PB 4
#define BUFW (MT * LDSPITCH)  // floats per buffer

__global__ void __launch_bounds__(128) hmm_emit(const float* __restrict__ z,
                                                const float* __restrict__ ws,
                                                float* __restrict__ le) {
    extern __shared__ float zt[];   // 2 * BUFW floats (dynamic: 66,560 B)
    const float* W1p  = ws + W1P_OFF;
    const float* W2p  = ws + W2P_OFF;
    const float* offk = ws + OFF_OFF;

    const int tid  = threadIdx.x;
    const int wave = tid >> 5;
    const int lane = tid & 31;
    const int mrow  = (wave & 1) * 16;
    const int n0    = (wave >> 1) * 16;
    const int lhalf = lane >> 4;
    const int l15   = lane & 15;
    const int row   = mrow + l15;          // A-matrix M index
    const int kcol  = n0 + l15;            // B/C/D N index (HMM state)
    const float off = offk[kcol];

#if HAVE_ASYNC_LDS
    // Issue one 32x256 tile as 16 async B128 per thread (ASYNCcnt += 16/wave).
    auto issue = [&](int tile, int buf) {
        const float* gbase = z + ((long)blockIdx.x * TILES_PB + tile) * (MT * DD);
        float* lbase = &zt[buf * BUFW];
        #pragma unroll
        for (int u = 0; u < 16; ++u) {
            int i  = tid + u * 128;       // float4 index within tile
            int r  = i >> 6;
            int c4 = i & 63;
            __builtin_amdgcn_global_load_async_to_lds_b128(
                to_gbl(gbase + i * 4), to_lds(&lbase[r * LDSPITCH + c4 * 4]),
                0, 0);
        }
    };
    issue(0, 0);
#endif

    for (int it = 0; it < TILES_PB; ++it) {
        const int buf = it & 1;
#if HAVE_ASYNC_LDS
        if (it + 1 < TILES_PB) {
            issue(it + 1, buf ^ 1);
            WAIT_ASYNC(16);   // in-order: previous tile's 16 loads done
        } else {
            WAIT_ASYNC(0);
        }
        __syncthreads();      // tile `it` visible to all waves
#else
        // Synchronous fallback: coalesced float4 fill.
        {
            const float* gbase =
                z + ((long)blockIdx.x * TILES_PB + it) * (MT * DD);
            float* lbase = &zt[buf * BUFW];
            for (int i = tid; i < MT * (DD / 4); i += 128) {
                int r  = i >> 6;
                int c4 = i & 63;
                float4 v = ((const float4*)gbase)[i];
                *(float4*)&lbase[r * LDSPITCH + c4 * 4] = v;
            }
        }
        __syncthreads();
#endif

        const float* arow = &zt[buf * BUFW + row * LDSPITCH + lhalf * 2];
        v8f cz2 = {};   // sum_d z^2 * iv
        v8f czm = {};   // sum_d z * (mu*iv)

        #pragma unroll 4
        for (int d0 = 0; d0 < DD; d0 += 4) {
            v2f a  = *(const v2f*)(arow + d0);
            v2f a2 = a * a;
            int P  = (d0 >> 1) + lhalf;
            v2f b1 = *(const v2f*)(W1p + (P * 32 + kcol) * 2);
            v2f b2 = *(const v2f*)(W2p + (P * 32 + kcol) * 2);
            cz2 = __builtin_amdgcn_wmma_f32_16x16x4_f32(false, a2, false, b1,
                                                        (short)0, cz2, false, false);
            czm = __builtin_amdgcn_wmma_f32_16x16x4_f32(false, a, false, b2,
                                                        (short)0, czm, false, false);
        }

        const long outbase = ((long)blockIdx.x * TILES_PB + it) * (MT * KK);
        // C/D layout: VGPR i -> M = i + 8*lhalf, N = l15
        #pragma unroll
        for (int i = 0; i < 8; ++i) {
            int m = mrow + i + 8 * lhalf;
            le[outbase + (long)m * KK + kcol] =
                (off - 0.5f * cz2[i] + czm[i]) * LOG2E;
        }
        __syncthreads();      // done reading buf before it is refilled
    }
}

// ---------------------------------------------------------------------------
// Forward scan in log2 domain: 1 wave32 per batch, one state per lane.
//   alpha_k <- le2[t,k] + log2sumexp2_j(alpha_j + lA2[j,k])
// exp2f/log2f map directly to v_exp_f32 / v_log_f32 (no scale mul).
// ---------------------------------------------------------------------------
__global__ void __launch_bounds__(32) hmm_scan(const float* __restrict__ ws,
                                               const float* __restrict__ le,
                                               float* __restrict__ out) {
    __shared__ float lAT[32 * 33];
    const int lane = threadIdx.x;
    const float* lpi = ws + LPI_OFF;
    const float* lA  = ws + LA_OFF;

    for (int e = lane; e < 1024; e += 32) {
        int j = e >> 5, k = e & 31;
        lAT[k * 33 + j] = lA[j * 32 + k];
    }
    __syncthreads();

    float rowA[32];
    #pragma unroll
    for (int j = 0; j < 32; ++j) rowA[j] = lAT[lane * 33 + j];

    const float* leb = le + (long)blockIdx.x * TT * KK;
    float alpha = lpi[lane] + leb[lane];

    for (int t = 1; t < TT; ++t) {
        if (t + 16 < TT)
            __builtin_prefetch(&leb[(t + 16) * KK + lane], 0, 0);
        float lev = leb[t * KK + lane];
        float v[32];
        #pragma unroll
        for (int j = 0; j < 32; ++j)
            v[j] = __shfl(alpha, j, 32) + rowA[j];
        float m = v[0];
        #pragma unroll
        for (int j = 1; j < 32; ++j) m = fmaxf(m, v[j]);
        float s = 0.0f;
        #pragma unroll
        for (int j = 0; j < 32; ++j) s += exp2f(v[j] - m);
        alpha = lev + m + log2f(s);
    }

    // log2-sum-exp2 across the 32 states; convert back to nats at the end.
    float m = alpha;
    #pragma unroll
    for (int o = 16; o > 0; o >>= 1) m = fmaxf(m, __shfl_xor(m, o, 32));
    float s = exp2f(alpha - m);
    #pragma unroll
    for (int o = 16; o > 0; o >>= 1) s += __shfl_xor(s, o, 32);
    float ll = (m + log2f(s)) * LN2;
    if (lane == 0) atomicAdd(out, -ll * (1.0f / (float)BB));
}

// ---------------------------------------------------------------------------
extern "C" void kernel_launch(void* const* d_in, const int* in_sizes, int n_in,
                              void* d_out, int out_size, void* d_ws, size_t ws_size,
                              hipStream_t stream) {
    const float* z           = (const float*)d_in[0];
    const float* init_logits = (const float*)d_in[1];
    const float* trans       = (const float*)d_in[2];
    const float* means       = (const float*)d_in[3];
    const float* log_vars    = (const float*)d_in[4];
    float* out = (float*)d_out;
    float* ws  = (float*)d_ws;
    float* le  = ws + LE_OFF;

    hmm_zero_out<<<1, 1, 0, stream>>>(out);
    hmm_prep<<<1, 256, 0, stream>>>(init_logits, trans, means, log_vars, ws);
    const size_t lds_bytes = 2u * BUFW * sizeof(float);  // 66,560 B (<320 KB/WGP)
    hmm_emit<<<(BB * TT) / (MT * TILES_PB), 128, lds_bytes, stream>>>(z, ws, le);
    hmm_scan<<<BB, 32, 0, stream>>>(ws, le, out);
}